// UnitCellThlista_78975858639047
// MI455X (gfx1250) — compile-verified
//
#include <hip/hip_runtime.h>
#include <math.h>

#define N_DIM 4096
#define M_DIM 2048
#define B_DIM 512
#define BETA_F 0.01f

typedef float v2f __attribute__((ext_vector_type(2)));
typedef float v8f __attribute__((ext_vector_type(8)));

// D = A(16x4 f32) * B(4x16 f32) + C(16x16 f32), fp32 WMMA (CDNA5)
__device__ __forceinline__ v8f wmma4(v2f a, v2f b, v8f c) {
  return __builtin_amdgcn_wmma_f32_16x16x4_f32(
      /*neg_a=*/false, a, /*neg_b=*/false, b,
      /*c_mod=*/(short)0, c, /*reuse_a=*/false, /*reuse_b=*/false);
}

// z = W1 @ x + W2 @ y, complex arithmetic folded into doubled-K real WMMA:
//   Re(z): A = (re W, -im W) per K-pair, B = (re u, im u) = native complex64 layout
//   Im(z): A = (im W,  re W) against the same B fragment.
// Phase-1 A fragments are pre-transformed into LDS (v1a = Re-form, v1b = Im-form,
// Toeplitz conjugation folded in) so the inner loop does zero fragment VALU work.
// Each wave: 2 row-tiles x 4 col-tiles = 32 rows x 64 cols of output.
__global__ __launch_bounds__(128)
void toeplitz_cgemm_soft_kernel(const float2* __restrict__ v,
                                const v2f* __restrict__ W2,
                                const v2f* __restrict__ x,
                                const v2f* __restrict__ y,
                                float2* __restrict__ out) {
  // W1[i,k] = v1[(N-1) + i - k], v1 = concat(conj(v[::-1]), v[1:])
  __shared__ v2f v1a[2 * N_DIM];  // (re W1, -im W1)  : Re-output A fragments, 64 KB
  __shared__ v2f v1b[2 * N_DIM];  // (im W1,  re W1)  : Im-output A fragments, 64 KB

  const int tid = threadIdx.x;
  for (int idx = tid; idx < 2 * N_DIM - 1; idx += 128) {
    const int d = idx - (N_DIM - 1);
    const int ad = d < 0 ? -d : d;
    const float2 t = v[ad];
    const float re = t.x;
    const float im = (d < 0) ? -t.y : t.y;  // conj above the diagonal
    v2f a; a.x = re;  a.y = -im;
    v2f b; b.x = im;  b.y = re;
    v1a[idx] = a;
    v1b[idx] = b;
  }
  __syncthreads();

  const int lane    = tid & 31;
  const int wid     = tid >> 5;
  const int rowBase = blockIdx.x * 128 + wid * 32;  // 2 row-tiles of 16
  const int colBase = blockIdx.y * 64;              // 4 col-tiles of 16
  const int mrow    = lane & 15;   // A row / B,C column within a 16-tile
  const int h       = lane >> 4;   // lane half -> which complex k of the pair
  const int i0      = rowBase + mrow;
  const int i1      = i0 + 16;

  v8f accR[2][4], accI[2][4];
#pragma unroll
  for (int rt = 0; rt < 2; ++rt)
#pragma unroll
    for (int t = 0; t < 4; ++t) {
      accR[rt][t] = (v8f)0.0f;
      accI[rt][t] = (v8f)0.0f;
    }

  // ---------------- Phase 1: W1 @ x (2 complex k per WMMA step) ----------------
  for (int kb = 0; kb < N_DIM; kb += 2) {
    const int k    = kb + h;
    const int idx0 = (N_DIM - 1) + i0 - k;
    const int idx1 = idx0 + 16;
    const v2f a0r = v1a[idx0];
    const v2f a0i = v1b[idx0];
    const v2f a1r = v1a[idx1];
    const v2f a1i = v1b[idx1];
#pragma unroll
    for (int t = 0; t < 4; ++t) {
      const v2f b = x[k * B_DIM + colBase + t * 16 + mrow];  // (re,im) = B fragment
      accR[0][t] = wmma4(a0r, b, accR[0][t]);
      accI[0][t] = wmma4(a0i, b, accI[0][t]);
      accR[1][t] = wmma4(a1r, b, accR[1][t]);
      accI[1][t] = wmma4(a1i, b, accI[1][t]);
    }
  }

  // ---------------- Phase 2: W2 @ y ----------------
  for (int kb = 0; kb < M_DIM; kb += 2) {
    const int k = kb + h;
    const v2f* w0p = &W2[(size_t)i0 * M_DIM + k];
    const v2f* w1p = &W2[(size_t)i1 * M_DIM + k];
    const v2f w0 = *w0p;
    const v2f w1 = *w1p;
    // warm lines ~16 complex k ahead (consumed over the next iterations)
    __builtin_prefetch(w0p + 16, 0, 1);
    __builtin_prefetch(w1p + 16, 0, 1);
    v2f a0r; a0r.x = w0.x; a0r.y = -w0.y;
    v2f a0i; a0i.x = w0.y; a0i.y = w0.x;
    v2f a1r; a1r.x = w1.x; a1r.y = -w1.y;
    v2f a1i; a1i.x = w1.y; a1i.y = w1.x;
#pragma unroll
    for (int t = 0; t < 4; ++t) {
      const v2f b = y[k * B_DIM + colBase + t * 16 + mrow];
      accR[0][t] = wmma4(a0r, b, accR[0][t]);
      accI[0][t] = wmma4(a0i, b, accI[0][t]);
      accR[1][t] = wmma4(a1r, b, accR[1][t]);
      accI[1][t] = wmma4(a1i, b, accI[1][t]);
    }
  }

  // ---------------- Epilogue: soft-threshold ----------------
  // C/D layout: VGPR j -> row M=j (lanes 0-15) or M=j+8 (lanes 16-31), col N=lane&15
  const int rsub = h * 8;
#pragma unroll
  for (int rt = 0; rt < 2; ++rt) {
#pragma unroll
    for (int t = 0; t < 4; ++t) {
      const int c = colBase + t * 16 + mrow;
#pragma unroll
      for (int j = 0; j < 8; ++j) {
        const float zr = accR[rt][t][j];
        const float zi = accI[rt][t][j];
        const float mag   = sqrtf(zr * zr + zi * zi);
        const float scale = fmaxf(mag - BETA_F, 0.0f) / fmaxf(mag, 1e-12f);
        const int r = rowBase + rt * 16 + rsub + j;
        out[(size_t)r * B_DIM + c] = make_float2(zr * scale, zi * scale);
      }
    }
  }
}

extern "C" void kernel_launch(void* const* d_in, const int* in_sizes, int n_in,
                              void* d_out, int out_size, void* d_ws, size_t ws_size,
                              hipStream_t stream) {
  const float2* v  = (const float2*)d_in[0];
  const v2f* W2    = (const v2f*)d_in[1];
  const v2f* x     = (const v2f*)d_in[2];
  const v2f* y     = (const v2f*)d_in[3];
  float2* out = (float2*)d_out;

  dim3 grid(N_DIM / 128, B_DIM / 64);  // 32 x 8 blocks
  dim3 block(128);                     // 4 waves of 32
  toeplitz_cgemm_soft_kernel<<<grid, block, 0, stream>>>(v, W2, x, y, out);
}